// YoloLoss_20315195310296
// MI455X (gfx1250) — compile-verified
//
#include <hip/hip_runtime.h>
#include <hip/hip_bf16.h>
#include <stdint.h>

// YOLO-v1 loss, algebraically reduced:
//   - IoU needed only at the center cell (I is masked by center_mask)
//   - noobject term = 0.25*(T - sum_{selected} pC^2), T precomputed per image
//   - class term uses per-cell S0 = sum_c p_c^2  (=> masked gather per object)
// One wave32 per image; predictions staged to LDS with CDNA5 async copies
// (IOFFSET applies to both LDS and global address -> one addr pair, 23 insts).

#define CELLS      49      // 7x7
#define CHS        30      // 20 class + 2 conf + 8 box
#define PRED_DW    1470    // 49*30 dwords per image
#define WAVES_PB   8
#define LDS_STRIDE 1536    // dwords per wave slice: 1470 pred + 49 S0 + pad
#define NIMG       4096
#define MAXOBJ     20

// one async b64 chunk per lane at immediate offset OFF (bytes, same for LDS+global)
#define ASYNC_B64(OFF)                                                  \
    asm volatile("global_load_async_to_lds_b64 %0, %1, off offset:" #OFF \
                 :: "v"(la), "v"(g) : "memory");

__global__ __launch_bounds__(256) void yolo_loss_kernel(
    const float* __restrict__ predicts,
    const float* __restrict__ labels,
    const int*   __restrict__ objects_num,
    float*       __restrict__ out)
{
    __shared__ float lds[WAVES_PB * LDS_STRIDE];
    __shared__ float partials[WAVES_PB];

    const int lane = threadIdx.x & 31;
    const int wave = threadIdx.x >> 5;
    // wave-uniform image index; readfirstlane lets uniform label/objects loads
    // take the scalar (SMEM) path
    const int b = __builtin_amdgcn_readfirstlane((int)(blockIdx.x * WAVES_PB + wave));

    float* wlds = &lds[wave * LDS_STRIDE];
    const float* pred = predicts + (size_t)b * PRED_DW;

    // ---- Stage predictions: global -> LDS via CDNA5 async copy (ASYNCcnt) ----
    // 735 b64 chunks; lane L owns chunks L, L+32, ... ; chunk stride/round = 256 B.
    {
        const unsigned la = (unsigned)(uintptr_t)(const void*)&wlds[lane * 2];
        const float*   g  = pred + lane * 2;
        ASYNC_B64(0)    ASYNC_B64(256)  ASYNC_B64(512)  ASYNC_B64(768)
        ASYNC_B64(1024) ASYNC_B64(1280) ASYNC_B64(1536) ASYNC_B64(1792)
        ASYNC_B64(2048) ASYNC_B64(2304) ASYNC_B64(2560) ASYNC_B64(2816)
        ASYNC_B64(3072) ASYNC_B64(3328) ASYNC_B64(3584) ASYNC_B64(3840)
        ASYNC_B64(4096) ASYNC_B64(4352) ASYNC_B64(4608) ASYNC_B64(4864)
        ASYNC_B64(5120) ASYNC_B64(5376)
        if (lane < 31) { ASYNC_B64(5632) }   // chunks 704..734
    }
    asm volatile("s_wait_asynccnt 0" ::: "memory");

    // ---- Phase A: per-image precompute (S0 per cell, T = sum pC^2) ----
    const int   c0   = lane;            // first owned cell (always < 49)
    const int   c1   = lane + 32;       // second owned cell if lane < 17
    const bool  has1 = (lane < 17);
    const float gi0 = (float)(c0 / 7), gj0 = (float)(c0 % 7);
    const float gi1 = (float)(c1 / 7), gj1 = (float)(c1 % 7);

    float tloc = 0.0f;
    {
        const float* cb = &wlds[c0 * CHS];
        float s = 0.0f;
        #pragma unroll
        for (int ch = 0; ch < 20; ++ch) { float v = cb[ch]; s += v * v; }
        wlds[PRED_DW + c0] = s;
        float a = cb[20], c = cb[21];
        tloc += a * a + c * c;
    }
    if (has1) {
        const float* cb = &wlds[c1 * CHS];
        float s = 0.0f;
        #pragma unroll
        for (int ch = 0; ch < 20; ++ch) { float v = cb[ch]; s += v * v; }
        wlds[PRED_DW + c1] = s;
        float a = cb[20], c = cb[21];
        tloc += a * a + c * c;
    }
    #pragma unroll
    for (int off = 16; off; off >>= 1) tloc += __shfl_xor(tloc, off, 32);
    const float T = tloc;   // uniform across the wave

    // ---- Phase B: object loop ----
    const float INV_GS = 1.0f / 64.0f;
    float acc  = 0.0f;   // lane-distributed class-loss part
    float accU = 0.0f;   // wave-uniform center/noobject part

    int n = objects_num[b];
    n = n < 0 ? 0 : (n > MAXOBJ ? MAXOBJ : n);
    const float* lab = labels + (size_t)b * MAXOBJ * 5;

    for (int m = 0; m < n; ++m) {
        const float lx = lab[m * 5 + 0];
        const float ly = lab[m * 5 + 1];
        const float lw = lab[m * 5 + 2];
        const float lh = lab[m * 5 + 3];
        const int icls = (int)lab[m * 5 + 4];

        const float minx = floorf((lx - 0.5f * lw) * INV_GS);
        const float maxx = fminf(ceilf((lx + 0.5f * lw) * INV_GS), 7.0f);
        const float miny = floorf((ly - 0.5f * lh) * INV_GS);
        const float maxy = fminf(ceilf((ly + 0.5f * lh) * INV_GS), 7.0f);

        // class loss over object_mask cells:  S0 - 2*p_cls + 1  (CLASS_SCALE*0.5 = 1)
        if (gj0 >= minx && gj0 < maxx && gi0 >= miny && gi0 < maxy)
            acc += wlds[PRED_DW + c0] - 2.0f * wlds[c0 * CHS + icls] + 1.0f;
        if (has1 && gj1 >= minx && gj1 < maxx && gi1 >= miny && gi1 < maxy)
            acc += wlds[PRED_DW + c1] - 2.0f * wlds[c1 * CHS + icls] + 1.0f;

        // center cell (uniform across lanes; LDS broadcast reads)
        const int cx = (int)floorf(lx * INV_GS);
        const int cy = (int)floorf(ly * INV_GS);
        const float* cc = &wlds[(cy * 7 + cx) * CHS];
        const float pC[2] = { cc[20], cc[21] };

        const float tx1 = lx - 0.5f * lw, ty1 = ly - 0.5f * lh;
        const float tx2 = lx + 0.5f * lw, ty2 = ly + 0.5f * lh;
        const float sq2 = (tx2 - tx1) * (ty2 - ty1);

        float iou[2], px[2], py[2], pw[2], ph[2];
        #pragma unroll
        for (int k = 0; k < 2; ++k) {
            const float rx = cc[22 + 4 * k], ry = cc[23 + 4 * k];
            const float rw = cc[24 + 4 * k], rh = cc[25 + 4 * k];
            px[k] = rx * 64.0f + 64.0f * (float)cx;
            py[k] = ry * 64.0f + 64.0f * (float)cy;
            pw[k] = rw * 448.0f;
            ph[k] = rh * 448.0f;
            const float x1 = px[k] - 0.5f * pw[k], y1 = py[k] - 0.5f * ph[k];
            const float x2 = px[k] + 0.5f * pw[k], y2 = py[k] + 0.5f * ph[k];
            float ix = fminf(x2, tx2) - fmaxf(x1, tx1);
            float iy = fminf(y2, ty2) - fmaxf(y1, ty1);
            ix = fmaxf(ix, 0.0f); iy = fmaxf(iy, 0.0f);
            const float inter = ix * iy;
            const float sq1 = (x2 - x1) * (y2 - y1);
            iou[k] = inter / (sq1 + sq2 - inter + 1e-6f);
        }
        const float mxiou = fmaxf(iou[0], iou[1]);
        const float sw = sqrtf(fabsf(lw)), sh = sqrtf(fabsf(lh));

        #pragma unroll
        for (int k = 0; k < 2; ++k) {
            if (iou[k] >= mxiou) {   // I_k == 1
                const float d = pC[k] - iou[k];
                accU += 0.5f * d * d;            // object loss (OBJECT_SCALE=1)
                accU -= 0.25f * pC[k] * pC[k];   // remove from noobject sum
                const float dx = (px[k] - lx) * INV_GS;
                const float dy = (py[k] - ly) * INV_GS;
                const float psw = sqrtf(fminf(fmaxf(pw[k], 0.0f), 448.0f));
                const float psh = sqrtf(fminf(fmaxf(ph[k], 0.0f), 448.0f));
                const float dw = psw - sw, dh = psh - sh;
                accU += 2.5f * (dx * dx + dy * dy)
                      + (2.5f / 448.0f) * (dw * dw + dh * dh);  // coord (5*0.5)
            }
        }
        accU += 0.25f * T;   // noobject base: 0.5*0.5 * sum_all pC^2
    }

    // ---- Wave reduce, then block reduce -> one atomic per block ----
    #pragma unroll
    for (int off = 16; off; off >>= 1) acc += __shfl_xor(acc, off, 32);
    if (lane == 0) partials[wave] = acc + accU;
    __syncthreads();
    if (threadIdx.x == 0) {
        float s = 0.0f;
        #pragma unroll
        for (int w = 0; w < WAVES_PB; ++w) s += partials[w];
        atomicAdd(out, s * (1.0f / (float)NIMG));
    }
}

extern "C" void kernel_launch(void* const* d_in, const int* in_sizes, int n_in,
                              void* d_out, int out_size, void* d_ws, size_t ws_size,
                              hipStream_t stream) {
    (void)in_sizes; (void)n_in; (void)out_size; (void)d_ws; (void)ws_size;
    const float* predicts    = (const float*)d_in[0];
    const float* labels      = (const float*)d_in[1];
    const int*   objects_num = (const int*)d_in[2];
    float* out = (float*)d_out;

    hipMemsetAsync(out, 0, sizeof(float), stream);
    dim3 grid(NIMG / WAVES_PB);   // 512 blocks
    dim3 block(32 * WAVES_PB);    // 256 threads = 8 wave32s
    yolo_loss_kernel<<<grid, block, 0, stream>>>(predicts, labels, objects_num, out);
}